// GaussianRasterizer_57724360458254
// MI455X (gfx1250) — compile-verified
//
#include <hip/hip_runtime.h>
#include <math.h>

#define NG     2048
#define IMG_W  128
#define IMG_H  128

typedef __attribute__((ext_vector_type(2))) float v2f;
typedef __attribute__((ext_vector_type(8))) float v8f;

// ------------------------------------------------------------------
// CDNA5 async global->LDS staging (ASYNCcnt-tracked), with fallback.
// Probe round 2 revealed the builtin's param types: non-const int4*
// in AS(1) (printed as "__device__") for the global source.
// ------------------------------------------------------------------
typedef int v4i __attribute__((vector_size(16)));
typedef __attribute__((address_space(1))) v4i* gv4p;   // global int4*
typedef __attribute__((address_space(3))) v4i* lv4p;   // LDS int4*

#if __has_builtin(__builtin_amdgcn_global_load_async_to_lds_b128)
#define HAVE_ASYNC_LDS 1
__device__ __forceinline__ void stage16(float4* lds_dst, const float4* gsrc) {
  // generic->as(1): same 64-bit value; generic->as(3): low 32 bits are the
  // LDS offset (ISA flat->LDS mapping discards upper bits).
  __builtin_amdgcn_global_load_async_to_lds_b128(
      (gv4p)(uintptr_t)gsrc,
      (lv4p)(unsigned)(uintptr_t)lds_dst,
      0, 0);
}
#else
#define HAVE_ASYNC_LDS 0
__device__ __forceinline__ void stage16(float4* lds_dst, const float4* gsrc) {
  *lds_dst = *gsrc;
}
#endif

__device__ __forceinline__ void wait_async0() {
#if HAVE_ASYNC_LDS
#if __has_builtin(__builtin_amdgcn_s_wait_asynccnt)
  __builtin_amdgcn_s_wait_asynccnt(0);
#else
  asm volatile("s_wait_asynccnt 0" ::: "memory");
#endif
#endif
}

// Param record per gaussian: 3 x float4
//  [0]: mux, muy, ic00, ic01
//  [1]: ic11, op,  radius, depth
//  [2]: colR, colG, colB, 0
// ------------------------------------------------------------------
// Kernel 1: per-gaussian preprocess (quat -> cov3d -> 2D conic)
// ------------------------------------------------------------------
__global__ __launch_bounds__(256) void gs_preprocess(
    const float* __restrict__ pos, const float* __restrict__ scl,
    const float* __restrict__ rot, const float* __restrict__ col,
    const float* __restrict__ opa, const float* __restrict__ view,
    float* __restrict__ depths, float* __restrict__ up) {
  int i = blockIdx.x * blockDim.x + threadIdx.x;
  if (i >= NG) return;

  float qw = rot[i*4+0], qx = rot[i*4+1], qy = rot[i*4+2], qz = rot[i*4+3];
  float nrm = sqrtf(qw*qw + qx*qx + qy*qy + qz*qz) + 1e-8f;
  float w = qw/nrm, x = qx/nrm, y = qy/nrm, z = qz/nrm;

  float R[3][3];
  R[0][0] = 1.f - 2.f*(y*y + z*z); R[0][1] = 2.f*(x*y + z*w); R[0][2] = 2.f*(x*z - y*w);
  R[1][0] = 2.f*(x*y - z*w); R[1][1] = 1.f - 2.f*(x*x + z*z); R[1][2] = 2.f*(y*z + x*w);
  R[2][0] = 2.f*(x*z + y*w); R[2][1] = 2.f*(y*z - x*w); R[2][2] = 1.f - 2.f*(x*x + y*y);

  float s2[3];
  #pragma unroll
  for (int j = 0; j < 3; ++j) {
    float s = fmaxf(scl[i*3+j], 1e-6f);
    s2[j] = s * s;
  }
  float cov[3][3];
  #pragma unroll
  for (int a = 0; a < 3; ++a)
    #pragma unroll
    for (int b = 0; b < 3; ++b) {
      float acc = 0.f;
      #pragma unroll
      for (int j = 0; j < 3; ++j) acc += R[a][j] * s2[j] * R[b][j];
      cov[a][b] = acc;
    }

  float Rv[3][3], tv[3];
  #pragma unroll
  for (int r = 0; r < 3; ++r) {
    #pragma unroll
    for (int c = 0; c < 3; ++c) Rv[r][c] = view[r*4 + c];
    tv[r] = view[r*4 + 3];
  }
  float p0 = pos[i*3+0], p1 = pos[i*3+1], p2 = pos[i*3+2];
  float X = Rv[0][0]*p0 + Rv[0][1]*p1 + Rv[0][2]*p2 + tv[0];
  float Y = Rv[1][0]*p0 + Rv[1][1]*p1 + Rv[1][2]*p2 + tv[1];
  float Z = Rv[2][0]*p0 + Rv[2][1]*p1 + Rv[2][2]*p2 + tv[2];
  float valid = (Z > 0.1f) ? 1.f : 0.f;

  const float FX = 110.85125168440814f;  // (W/2)/tan(30deg)
  float Zs = fmaxf(Z, 1e-6f);
  float mux = FX * X / Zs + 64.f;
  float muy = FX * Y / Zs + 64.f;
  float invZ2 = 1.f / (Zs * Zs);
  float J0[3] = { FX / Zs, 0.f, -FX * X * invZ2 };
  float J1[3] = { 0.f, FX / Zs, -FX * Y * invZ2 };

  float SRt0[3], SRt1[3];
  #pragma unroll
  for (int a = 0; a < 3; ++a) {
    SRt0[a] = cov[a][0]*Rv[0][0] + cov[a][1]*Rv[0][1] + cov[a][2]*Rv[0][2];
    SRt1[a] = cov[a][0]*Rv[1][0] + cov[a][1]*Rv[1][1] + cov[a][2]*Rv[1][2];
  }
  float c00 = J0[0]*SRt0[0] + J0[1]*SRt0[1] + J0[2]*SRt0[2];
  float c01 = J0[0]*SRt1[0] + J0[1]*SRt1[1] + J0[2]*SRt1[2];
  float c11 = J1[0]*SRt1[0] + J1[1]*SRt1[1] + J1[2]*SRt1[2];

  float tr  = c00 + c11;
  float det = c00 * c11 - c01 * c01;
  float disc = fmaxf(tr*tr - 4.f*det, 0.f);
  float lam  = 0.5f * (tr + sqrtf(disc));
  float radius = 3.f * sqrtf(fmaxf(lam, 1e-8f));
  c00 += 1e-6f;
  c11 += 1e-6f;

  float det2 = c00*c11 - c01*c01 + 1e-8f;
  float ic00 =  c11 / det2;
  float ic01 = -c01 / det2;
  float ic11 =  c00 / det2;
  float op   = opa[i] * valid;

  depths[i] = Z;
  float* o = up + i*12;
  o[0] = mux;  o[1] = muy;  o[2]  = ic00;      o[3]  = ic01;
  o[4] = ic11; o[5] = op;   o[6]  = radius;    o[7]  = Z;
  o[8] = col[i*3+0]; o[9] = col[i*3+1]; o[10] = col[i*3+2]; o[11] = 0.f;
}

// ------------------------------------------------------------------
// Kernel 2: stable rank by descending depth (== argsort(-depths)),
// scatter params into sorted order.
// ------------------------------------------------------------------
__global__ __launch_bounds__(256) void gs_rank_scatter(
    const float* __restrict__ depths, const float* __restrict__ up,
    float* __restrict__ sp) {
  int i = blockIdx.x * blockDim.x + threadIdx.x;
  if (i >= NG) return;
  float di = depths[i];
  int rank = 0;
  for (int j = 0; j < NG; ++j) {
    float dj = depths[j];
    rank += (dj > di) || (dj == di && j < i);
  }
  #pragma unroll
  for (int k = 0; k < 12; ++k) sp[rank*12 + k] = up[i*12 + k];
}

// ------------------------------------------------------------------
// Kernel 3: render. One wave = 16 pixels, one block = one image row.
// Sorted gaussian table async-staged into LDS (96KB of 320KB WGP LDS).
// weight x color accumulated with V_WMMA_F32_16X16X4_F32.
// ------------------------------------------------------------------
__device__ __forceinline__ float eval_alpha(float4 a, float4 b, float px, float py) {
  float dx = px - a.x, dy = py - a.y;
  float e = -0.5f * (a.z*dx*dx + 2.f*a.w*dx*dy + b.x*dy*dy);
  e = fminf(e, 0.f);
  float r = b.z;
  float pxmin = fmaxf(0.f, truncf(a.x - r));
  float pxmax = fminf((float)IMG_W, truncf(a.x + r) + 1.f);
  float pymin = fmaxf(0.f, truncf(a.y - r));
  float pymax = fminf((float)IMG_H, truncf(a.y + r) + 1.f);
  bool inb = (px >= pxmin) && (px < pxmax) && (py >= pymin) && (py < pymax);
  return inb ? b.y * __expf(e) : 0.f;
}

__global__ __launch_bounds__(256) void gs_render(
    const float* __restrict__ sp, float* __restrict__ out) {
  __shared__ float4 sh[NG * 3];  // 98304 bytes

  const float4* gp = (const float4*)sp;
  for (int t = threadIdx.x; t < NG * 3; t += blockDim.x) stage16(&sh[t], &gp[t]);
  wait_async0();
  __syncthreads();

  int lane = threadIdx.x & 31;
  int wave = threadIdx.x >> 5;
  int m    = lane & 15;    // pixel within tile == WMMA row M == WMMA col N
  int half = lane >> 4;    // K-pair selector within chunk of 4

  float px = (float)(wave * 16 + m);
  float py = (float)blockIdx.x;

  float T = 1.f;
  v8f acc = {};

  // 64 super-chunks of 32 gaussians; inner 8 chunks of 4 fully unrolled
  // so the loop body carries 8 static v_wmma instructions.
  for (int sch = 0; sch < NG / 32; ++sch) {
    #pragma unroll
    for (int cc = 0; cc < 8; ++cc) {
      int c  = sch * 8 + cc;
      int g0 = 4*c + 2*half;
      float4 a0 = sh[g0*3 + 0],     b0 = sh[g0*3 + 1],     col0 = sh[g0*3 + 2];
      float4 a1 = sh[(g0+1)*3 + 0], b1 = sh[(g0+1)*3 + 1], col1 = sh[(g0+1)*3 + 2];

      float al0 = eval_alpha(a0, b0, px, py);
      float al1 = eval_alpha(a1, b1, px, py);

      float p  = (1.f - al0) * (1.f - al1);   // this half's transmittance factor
      float q  = __shfl_xor(p, 16, 32);       // other half's factor
      float Tb = half ? (T * q) : T;          // K=2,3 come after K=0,1

      v2f A;                                  // weights: A[M=m][K=2*half+{0,1}]
      A.x = al0 * Tb;
      A.y = al1 * Tb * (1.f - al0);

      v2f B;                                  // colors: B[K][N=m], N>=3 -> 0
      B.x = (m == 0) ? col0.x : (m == 1) ? col0.y : (m == 2) ? col0.z : 0.f;
      B.y = (m == 0) ? col1.x : (m == 1) ? col1.y : (m == 2) ? col1.z : 0.f;

      acc = __builtin_amdgcn_wmma_f32_16x16x4_f32(
          /*neg_a=*/false, A, /*neg_b=*/false, B,
          /*c_mod=*/(short)0, acc, /*reuse_a=*/false, /*reuse_b=*/false);

      T = T * p * q;
    }
    // wave-uniform early exit (EXEC stays all-1s around the WMMAs)
#if __has_builtin(__builtin_amdgcn_ballot_w32)
    if (__builtin_amdgcn_ballot_w32(T >= 1e-7f) == 0u) break;
#else
    if (__all(T < 1e-7f)) break;
#endif
  }

  // D[M=r+8*half][N=m] lives in acc[r] of lane (16*half + m). Channels are N=0..2.
  if (m < 3) {
    int base = ((int)blockIdx.x * IMG_W + wave * 16) * 3;
    #pragma unroll
    for (int r = 0; r < 8; ++r) {
      int pix = half * 8 + r;
      out[base + pix * 3 + m] = acc[r];
    }
  }
}

// ------------------------------------------------------------------
extern "C" void kernel_launch(void* const* d_in, const int* in_sizes, int n_in,
                              void* d_out, int out_size, void* d_ws, size_t ws_size,
                              hipStream_t stream) {
  (void)in_sizes; (void)n_in; (void)out_size; (void)ws_size;
  const float* pos  = (const float*)d_in[0];
  const float* scl  = (const float*)d_in[1];
  const float* rot  = (const float*)d_in[2];
  const float* col  = (const float*)d_in[3];
  const float* opa  = (const float*)d_in[4];
  const float* view = (const float*)d_in[5];

  float* ws     = (float*)d_ws;
  float* depths = ws;                 // NG floats
  float* up     = ws + NG;            // NG*12 floats (unsorted params)
  float* sp     = ws + NG + NG*12;    // NG*12 floats (sorted params, 16B aligned)

  gs_preprocess<<<NG/256, 256, 0, stream>>>(pos, scl, rot, col, opa, view, depths, up);
  gs_rank_scatter<<<NG/256, 256, 0, stream>>>(depths, up, sp);
  gs_render<<<IMG_H, 256, 0, stream>>>(sp, (float*)d_out);
}